// BaselineGraphSAGE_28346784153651
// MI455X (gfx1250) — compile-verified
//
#include <hip/hip_runtime.h>

typedef __attribute__((ext_vector_type(2))) float v2f;
typedef __attribute__((ext_vector_type(8))) float v8f;

#define NNODES 100000
#define NEDGES 1600000
#define INCH   128
#define HID    128
#define OUTCH  64

// ---------------------------------------------------------------- atomics
// Guaranteed single hardware instruction (no CAS-loop expansion).
// No-return form -> tracked by STOREcnt; S_ENDPGM's implicit wait-idle drains it.
__device__ __forceinline__ void hw_atomic_add_f32(float* p, float v) {
  asm volatile("global_atomic_add_f32 %0, %1, off" : : "v"(p), "v"(v) : "memory");
}

// ---------------------------------------------------------------- utilities
__global__ void zero_f32(float* __restrict__ p, int n) {
  int i = blockIdx.x * blockDim.x + threadIdx.x;
  if (i < n) p[i] = 0.0f;
}

__global__ void deg_count(const int* __restrict__ dst, float* __restrict__ deg, int nE) {
  int e = blockIdx.x * blockDim.x + threadIdx.x;
  if (e < nE) hw_atomic_add_f32(&deg[dst[e]], 1.0f);
}

__global__ void deg_to_inv(float* __restrict__ deg, int n) {
  int i = blockIdx.x * blockDim.x + threadIdx.x;
  if (i < n) {
    float d = deg[i];
    deg[i] = (d > 0.0f) ? (1.0f / d) : 0.0f;
  }
}

// W[K,N] row-major -> Wt[N,K] (N-major) so GEMM B loads are contiguous float2
__global__ void transpose_w(const float* __restrict__ W, float* __restrict__ Wt,
                            int K, int N) {
  int t = blockIdx.x * blockDim.x + threadIdx.x;
  if (t < K * N) {
    int k = t % K;
    int n = t / K;
    Wt[(size_t)n * K + k] = W[(size_t)k * N + n];
  }
}

// one thread per (edge, channel); wave = 32 consecutive channels -> coalesced
__global__ void scatter_add_feat(const float* __restrict__ feat,
                                 const int* __restrict__ src,
                                 const int* __restrict__ dst,
                                 float* __restrict__ agg, int nE) {
  long long tid = (long long)blockIdx.x * blockDim.x + threadIdx.x;
  int e = (int)(tid >> 7);
  int c = (int)(tid & 127);
  if (e < nE) {
    int s = src[e];
    int d = dst[e];
    float v = feat[(size_t)s * 128 + c];
    hw_atomic_add_f32(&agg[(size_t)d * 128 + c], v);
  }
}

// ---------------------------------------------------------------- WMMA GEMM
// out[m0:m0+16, :] = act( (agg*deg_inv) @ Wl + xin @ Wr + bias )
// K fixed = 128. One wave per 16-col N-tile; blockDim = (N/16)*32.
// Weights arrive pre-transposed: Wt[n*K + k]. Uses V_WMMA_F32_16X16X4_F32.
template <int N, bool RELU>
__global__ __launch_bounds__((N / 16) * 32)
void sage_gemm(const float* __restrict__ agg,      // [M,128] un-normalized sums
               const float* __restrict__ deg_inv,  // [M]
               const float* __restrict__ xin,      // [M,128] root features
               const float* __restrict__ Wlt,      // [N,128] transposed
               const float* __restrict__ Wrt,      // [N,128] transposed
               const float* __restrict__ bias,     // [N]
               float* __restrict__ out)            // [M,N]
{
  constexpr int K = 128;
  constexpr int KP = K + 4;  // LDS row pad: 32 lanes' float2 reads hit 64 banks
  __shared__ float sMean[16 * KP];
  __shared__ float sX[16 * KP];

  const int m0 = blockIdx.x * 16;
  const int lane = threadIdx.x & 31;
  const int wave = threadIdx.x >> 5;

  // stage A tiles (mean & x) once per block, shared by all N-tile waves
  for (int i = threadIdx.x; i < 16 * K; i += blockDim.x) {
    int r = i >> 7;           // row in tile
    int c = i & 127;          // K index
    size_t g = (size_t)(m0 + r) * K + c;
    float di = deg_inv[m0 + r];
    sMean[r * KP + c] = agg[g] * di;
    sX[r * KP + c]    = xin[g];
  }
  __syncthreads();

  // per-lane WMMA operand coordinates (A 16x4 f32 / B 4x16 f32 layouts)
  const int arow = lane & 15;           // A row (both lane halves cover M=0..15)
  const int koff = (lane >> 4) << 1;    // 0 for lanes 0-15, 2 for lanes 16-31
  const int n0 = wave * 16;
  const int bn = n0 + (lane & 15);      // B column this lane supplies
  const float* wl = Wlt + (size_t)bn * K;
  const float* wr = Wrt + (size_t)bn * K;

  v8f acc = {};
#pragma unroll
  for (int k = 0; k < K; k += 4) {
    v2f aM, aX;
    const float* pm = &sMean[arow * KP + k + koff];
    aM.x = pm[0]; aM.y = pm[1];
    const float* px = &sX[arow * KP + k + koff];
    aX.x = px[0]; aX.y = px[1];
    // contiguous 8B-aligned pairs from the transposed weights -> b64 loads
    v2f bL = *(const v2f*)(wl + k + koff);
    v2f bR = *(const v2f*)(wr + k + koff);
    // D = A*B + C   (8 args: neg_a, A, neg_b, B, c_mod, C, reuse_a, reuse_b)
    acc = __builtin_amdgcn_wmma_f32_16x16x4_f32(false, aM, false, bL,
                                                (short)0, acc, false, false);
    acc = __builtin_amdgcn_wmma_f32_16x16x4_f32(false, aX, false, bR,
                                                (short)0, acc, false, false);
  }

  // epilogue: C/D layout -> VGPR r holds row m0 + r (+8 for upper lane half)
  const int crow = m0 + ((lane >> 4) << 3);
  const float bv = bias[bn];
#pragma unroll
  for (int r = 0; r < 8; ++r) {
    float v = acc[r] + bv;
    if (RELU) v = fmaxf(v, 0.0f);
    out[(size_t)(crow + r) * N + bn] = v;
  }
}

// ---------------------------------------------------------------- launcher
extern "C" void kernel_launch(void* const* d_in, const int* in_sizes, int n_in,
                              void* d_out, int out_size, void* d_ws, size_t ws_size,
                              hipStream_t stream) {
  const float* x   = (const float*)d_in[0];
  const int*   ei  = (const int*)d_in[1];   // [2, NEDGES]: row0=src, row1=dst
  const float* W1l = (const float*)d_in[2];
  const float* W1r = (const float*)d_in[3];
  const float* b1  = (const float*)d_in[4];
  const float* W2l = (const float*)d_in[5];
  const float* W2r = (const float*)d_in[6];
  const float* b2  = (const float*)d_in[7];
  float* out = (float*)d_out;

  const int* src = ei;
  const int* dst = ei + NEDGES;

  // workspace layout (floats): deg | agg | h | W1lt | W1rt | W2lt | W2rt
  float* deg  = (float*)d_ws;
  float* agg  = deg + ((NNODES + 255) & ~255);
  float* h    = agg + (size_t)NNODES * HID;
  float* W1lt = h + (size_t)NNODES * HID;
  float* W1rt = W1lt + INCH * HID;
  float* W2lt = W1rt + INCH * HID;
  float* W2rt = W2lt + HID * OUTCH;

  const int TB = 256;
  const long long aggWork = (long long)NEDGES * 128;
  const int aggBlocks = (int)((aggWork + TB - 1) / TB);

  // transpose weights (tiny, L2-resident)
  transpose_w<<<(INCH * HID + TB - 1) / TB, TB, 0, stream>>>(W1l, W1lt, INCH, HID);
  transpose_w<<<(INCH * HID + TB - 1) / TB, TB, 0, stream>>>(W1r, W1rt, INCH, HID);
  transpose_w<<<(HID * OUTCH + TB - 1) / TB, TB, 0, stream>>>(W2l, W2lt, HID, OUTCH);
  transpose_w<<<(HID * OUTCH + TB - 1) / TB, TB, 0, stream>>>(W2r, W2rt, HID, OUTCH);

  // degree -> deg_inv (stored in-place in deg)
  zero_f32<<<(NNODES + TB - 1) / TB, TB, 0, stream>>>(deg, NNODES);
  zero_f32<<<(NNODES * HID + TB - 1) / TB, TB, 0, stream>>>(agg, NNODES * HID);
  deg_count<<<(NEDGES + TB - 1) / TB, TB, 0, stream>>>(dst, deg, NEDGES);
  deg_to_inv<<<(NNODES + TB - 1) / TB, TB, 0, stream>>>(deg, NNODES);

  // layer 1: aggregate x, then h = relu(mean@W1_l + x@W1_r + b1)
  scatter_add_feat<<<aggBlocks, TB, 0, stream>>>(x, src, dst, agg, NEDGES);
  sage_gemm<HID, true><<<NNODES / 16, (HID / 16) * 32, 0, stream>>>(
      agg, deg, x, W1lt, W1rt, b1, h);

  // layer 2: aggregate h, then out = mean2@W2_l + h@W2_r + b2
  zero_f32<<<(NNODES * HID + TB - 1) / TB, TB, 0, stream>>>(agg, NNODES * HID);
  scatter_add_feat<<<aggBlocks, TB, 0, stream>>>(h, src, dst, agg, NEDGES);
  sage_gemm<OUTCH, false><<<NNODES / 16, (OUTCH / 16) * 32, 0, stream>>>(
      agg, deg, h, W2lt, W2rt, b2, out);
}